// Edge_Cycle_50869592655552
// MI455X (gfx1250) — compile-verified
//
#include <hip/hip_runtime.h>
#include <hip/hip_bf16.h>
#include <cstddef>

// ---------------------------------------------------------------------------
// Problem constants (match reference)
// ---------------------------------------------------------------------------
#define H 64
#define NNODES 40000
#define NE 60000
#define NC5 30000
#define NC6 25000
#define E_ROWS (2 * NE)       // 120000
#define C5_ROWS (5 * NC5)     // 150000
#define C6_ROWS (6 * NC6)     // 150000
#define BN_EPS 1e-5f
#define MT 3                  // row-tiles (of 16) per block: 48 | 120000, 150000

typedef __attribute__((ext_vector_type(2))) float v2f;
typedef __attribute__((ext_vector_type(8))) float v8f;

// ---------------------------------------------------------------------------
// Workspace layout (float offsets)
// ---------------------------------------------------------------------------
// zeroed region (atomic accumulators + stats + counts):
#define OFF_NODE64    ((size_t)0)
#define OFF_NODE128_5 (OFF_NODE64 + (size_t)NNODES * 64)
#define OFF_NODE128_6 (OFF_NODE128_5 + (size_t)NNODES * 128)
#define OFF_NODE320   (OFF_NODE128_6 + (size_t)NNODES * 128)
#define OFF_STATS     (OFF_NODE320 + (size_t)NNODES * 320)  // 6 * 256
#define OFF_CNT5      (OFF_STATS + (size_t)6 * 256)         // 40000 ints
#define OFF_CNT6      (OFF_CNT5 + (size_t)NNODES)
#define ZERO_COUNT    (OFF_CNT6 + (size_t)NNODES)
// plain-write region:
#define OFF_BC64_5    (ZERO_COUNT)
#define OFF_BC64_6    (OFF_BC64_5 + (size_t)NC5 * 64)
#define OFF_BC128_5   (OFF_BC64_6 + (size_t)NC6 * 64)
#define OFF_BC128_6   (OFF_BC128_5 + (size_t)NC5 * 128)
#define OFF_H1        (OFF_BC128_6 + (size_t)NC6 * 128)    // reused: 150000*128
#define OFF_SB        (OFF_H1 + (size_t)C5_ROWS * 128)     // 6 * 256
// packed weights (WMMA B-fragment order):
#define OFF_WPK_E1    (OFF_SB + (size_t)6 * 256)           // 704*128
#define OFF_WPK_E2    (OFF_WPK_E1 + (size_t)704 * 128)     // 128*64
#define OFF_WPK_C1    (OFF_WPK_E2 + (size_t)128 * 64)      // 320*128
#define OFF_WPK_C2    (OFF_WPK_C1 + (size_t)320 * 128)     // 128*64

#define SET_E1 0
#define SET_E2 1
#define SET_C51 2
#define SET_C52 3
#define SET_C61 4
#define SET_C62 5

// d_out: edge_out [120000*64] | cycle5_out [150000*64] | cycle6_out [150000*64]
#define OUT_OFF_C5 ((size_t)E_ROWS * H)
#define OUT_OFF_C6 (OUT_OFF_C5 + (size_t)C5_ROWS * H)

// ---------------------------------------------------------------------------
// Utility kernels (float4-vectorized streaming)
// ---------------------------------------------------------------------------
__global__ void zero_f32x4(float4* __restrict__ p, size_t n4) {
  size_t i = (size_t)blockIdx.x * blockDim.x + threadIdx.x;
  size_t stride = (size_t)gridDim.x * blockDim.x;
  float4 z = {0.f, 0.f, 0.f, 0.f};
  for (; i < n4; i += stride) p[i] = z;
}

// repack W[K x N] row-major -> Wp[((k/4)*N + n)*4 + (k%4)]
// so a lane's (W[kk][n], W[kk+1][n]) pair is one aligned b64 load.
__global__ void pack_w(const float* __restrict__ W, float* __restrict__ Wp,
                       int K, int N) {
  int idx = blockIdx.x * blockDim.x + threadIdx.x;
  if (idx >= K * N) return;
  int k = idx / N, n = idx - k * N;
  Wp[(((size_t)(k >> 2) * N + n) << 2) + (k & 3)] = W[idx];
}

// cnt[a] += #rows with atom a
__global__ void count_atoms(const int* __restrict__ atoms, int rows,
                            int* __restrict__ cnt) {
  int r = blockIdx.x * blockDim.x + threadIdx.x;
  if (r >= rows) return;
  atomicAdd(&cnt[atoms[r]], 1);
}

// node64[a][k] += edge_feats[r][k]  (true scatter)
__global__ void scatter_node64(const float* __restrict__ src,
                               const int* __restrict__ atoms,
                               float* __restrict__ node, int rows) {
  size_t idx = (size_t)blockIdx.x * blockDim.x + threadIdx.x;
  int r = (int)(idx >> 6);
  int k = (int)(idx & 63);
  if (r >= rows) return;
  atomicAdd(&node[(size_t)atoms[r] * 64 + k], src[(size_t)r * 64 + k]);
}

// bc[c][k4] = sum_{i<S} node[atoms[c*S+i]][k4]   (float4 lanes, C4 = C/4)
__global__ void seg_sum_gather4(const float4* __restrict__ node,
                                const int* __restrict__ atoms,
                                float4* __restrict__ bc,
                                int nseg, int C4, int S) {
  size_t idx = (size_t)blockIdx.x * blockDim.x + threadIdx.x;
  int c = (int)(idx / C4);
  int k = (int)(idx % C4);
  if (c >= nseg) return;
  float4 s = {0.f, 0.f, 0.f, 0.f};
  for (int i = 0; i < S; ++i) {
    float4 v = node[(size_t)atoms[c * S + i] * C4 + k];
    s.x += v.x; s.y += v.y; s.z += v.z; s.w += v.w;
  }
  bc[(size_t)c * C4 + k] = s;
}

// node[a][base+k] += bc[r/S][k]
__global__ void scatter_bc(const float* __restrict__ bc,
                           const int* __restrict__ atoms,
                           float* __restrict__ node,
                           int rows, int S, int C, int ldN, int base) {
  size_t idx = (size_t)blockIdx.x * blockDim.x + threadIdx.x;
  int r = (int)(idx / C);
  int k = (int)(idx % C);
  if (r >= rows) return;
  atomicAdd(&node[(size_t)atoms[r] * ldN + base + k],
            bc[(size_t)(r / S) * C + k]);
}

// node320[a][256+k] += feats[r][k]
__global__ void scatter_feats(const float* __restrict__ feats,
                              const int* __restrict__ atoms,
                              float* __restrict__ node320, int rows) {
  size_t idx = (size_t)blockIdx.x * blockDim.x + threadIdx.x;
  int r = (int)(idx >> 6);
  int k = (int)(idx & 63);
  if (r >= rows) return;
  atomicAdd(&node320[(size_t)atoms[r] * 320 + 256 + k],
            feats[(size_t)r * 64 + k]);
}

// node128[a][k4] = cnt[a]*node64[a][k4]  (k4 < 16; node128 row = 32 float4)
__global__ void fill_node128_4(const float4* __restrict__ node64,
                               const int* __restrict__ cnt,
                               float4* __restrict__ node128) {
  size_t idx = (size_t)blockIdx.x * blockDim.x + threadIdx.x;
  int a = (int)(idx >> 4);
  int k = (int)(idx & 15);
  if (a >= NNODES) return;
  float c = (float)cnt[a];
  float4 v = node64[(size_t)a * 16 + k];
  float4 o = {c * v.x, c * v.y, c * v.z, c * v.w};
  node128[(size_t)a * 32 + k] = o;
}

// node320[a][k4] = cnt5[a]*n5[a][k4] + cnt6[a]*n6[a][k4]  (k4 < 32; row = 80)
__global__ void fill_node320_4(const float4* __restrict__ n5,
                               const float4* __restrict__ n6,
                               const int* __restrict__ cnt5,
                               const int* __restrict__ cnt6,
                               float4* __restrict__ node320) {
  size_t idx = (size_t)blockIdx.x * blockDim.x + threadIdx.x;
  int a = (int)(idx >> 5);
  int k = (int)(idx & 31);
  if (a >= NNODES) return;
  float p = (float)cnt5[a], q = (float)cnt6[a];
  float4 u = n5[(size_t)a * 32 + k];
  float4 v = n6[(size_t)a * 32 + k];
  float4 o = {p * u.x + q * v.x, p * u.y + q * v.y,
              p * u.z + q * v.z, p * u.w + q * v.w};
  node320[(size_t)a * 80 + k] = o;
}

// column sum / sumsq: stats[k] += sum, stats[C+k] += sumsq
__global__ void col_stats(const float* __restrict__ x, int R, int C,
                          float* __restrict__ stats) {
  int k = threadIdx.x;  // blockDim.x == C
  int rowsPer = (R + gridDim.x - 1) / gridDim.x;
  int r0 = blockIdx.x * rowsPer;
  int r1 = r0 + rowsPer; if (r1 > R) r1 = R;
  float s = 0.0f, s2 = 0.0f;
  for (int r = r0; r < r1; ++r) {
    float v = x[(size_t)r * C + k];
    s += v; s2 += v * v;
  }
  atomicAdd(&stats[k], s);
  atomicAdd(&stats[C + k], s2);
}

__global__ void bn_finalize(const float* __restrict__ stats,
                            const float* __restrict__ gamma,
                            const float* __restrict__ beta,
                            float* __restrict__ sb, int R, int C) {
  int k = threadIdx.x;
  if (k >= C) return;
  float mean = stats[k] / (float)R;
  float var = stats[C + k] / (float)R - mean * mean;
  if (var < 0.0f) var = 0.0f;
  float s = gamma[k] * rsqrtf(var + BN_EPS);
  sb[k] = s;
  sb[C + k] = beta[k] - mean * s;
}

// in-place x = relu(s*x + t), float4 over rows (C multiple of 4)
__global__ void bn_relu_apply4(float4* __restrict__ x,
                               const float* __restrict__ sb,
                               size_t n4, int C) {
  size_t idx = (size_t)blockIdx.x * blockDim.x + threadIdx.x;
  if (idx >= n4) return;
  int C4 = C >> 2;
  int k = ((int)(idx % C4)) << 2;
  float4 v = x[idx];
  v.x = fmaxf(sb[k + 0] * v.x + sb[C + k + 0], 0.0f);
  v.y = fmaxf(sb[k + 1] * v.y + sb[C + k + 1], 0.0f);
  v.z = fmaxf(sb[k + 2] * v.z + sb[C + k + 2], 0.0f);
  v.w = fmaxf(sb[k + 3] * v.w + sb[C + k + 3], 0.0f);
  x[idx] = v;
}

// ---------------------------------------------------------------------------
// WMMA GEMM: out[R x Nout] = A[R x K] @ W[K x Nout]
// A tile (48 x K) gather-assembled once per block into LDS; inner loop is
// ds_load_b64 (A) + one global_load_b64 (packed W) + 3x v_wmma_f32_16x16x4_f32.
// LDS row stride K+4 (K % 64 == 0) -> b64 A reads are bank-conflict-free.
// ---------------------------------------------------------------------------
struct GemmArgs {
  const float* Wp;     // packed [K/4][Nout][4]
  float* out;          // [R x Nout]
  int K, Nout;
  const float* feats;  // MODE0: edge_feats  MODE1: cycle feats  MODE2: raw h1
  const float* node;   // MODE0: node320     MODE1: node128
  const float* bcbuf;  // MODE1: bc128
  const int* atoms;    // MODE0/1
  const float* sb;     // MODE2: scale [0..K), shift [K..2K)
  int S;               // MODE1: rows per segment
};

#define GM_EDGE1 0
#define GM_CYC1 1
#define GM_BNRELU 2

template <int MODE>
__device__ __forceinline__ float gather_elem(const GemmArgs& g, int r, int k) {
  if constexpr (MODE == GM_EDGE1) {
    // [edge_feats(64) | node320[atom](320) | node320[a0]+node320[a1](320)]
    if (k < 64) return g.feats[(size_t)r * 64 + k];
    if (k < 384) return g.node[(size_t)g.atoms[r] * 320 + (k - 64)];
    int e = r >> 1;
    return g.node[(size_t)g.atoms[2 * e] * 320 + (k - 384)]
         + g.node[(size_t)g.atoms[2 * e + 1] * 320 + (k - 384)];
  } else if constexpr (MODE == GM_CYC1) {
    // [node128[atom](128) | bc128[r/S](128) | feats[r](64)]
    if (k < 128) return g.node[(size_t)g.atoms[r] * 128 + k];
    if (k < 256) return g.bcbuf[(size_t)(r / g.S) * 128 + (k - 128)];
    return g.feats[(size_t)r * 64 + (k - 256)];
  } else {
    // relu(s * raw + t)
    return fmaxf(g.sb[k] * g.feats[(size_t)r * g.K + k] + g.sb[g.K + k], 0.0f);
  }
}

template <int MODE>
__global__ __launch_bounds__(256) void gemm_wmma(GemmArgs g) {
  extern __shared__ float As[];  // [48][K+4]
  const int K = g.K;
  const int ldA = K + 4;
  const int m0 = blockIdx.x * (16 * MT);
  const int tid = threadIdx.x;
  const int nthr = blockDim.x;

  // cooperative gather-assembly of the A tile
  for (int idx = tid; idx < 16 * MT * K; idx += nthr) {
    int row = idx / K;
    int k = idx - row * K;
    As[row * ldA + k] = gather_elem<MODE>(g, m0 + row, k);
  }
  __syncthreads();

  const int wave = tid >> 5;              // wave32
  const int lane = tid & 31;
  const int ncol = wave * 16 + (lane & 15);
  const int koff = (lane < 16) ? 0 : 2;
  const int arow = lane & 15;
  const float* a0p = &As[(arow + 0) * ldA];
  const float* a1p = &As[(arow + 16) * ldA];
  const float* a2p = &As[(arow + 32) * ldA];
  // packed-W walk: one aligned b64 per k-step, constant stride Nout*4 floats
  const float* wp = &g.Wp[((size_t)ncol << 2) + koff];
  const int wstride = g.Nout << 2;

  v8f acc0 = {0.f, 0.f, 0.f, 0.f, 0.f, 0.f, 0.f, 0.f};
  v8f acc1 = acc0, acc2 = acc0;

#pragma unroll 4
  for (int k0 = 0; k0 < K; k0 += 4) {
    int kk = k0 + koff;
    v2f b = *(const v2f*)(wp + (size_t)(k0 >> 2) * wstride);
    v2f a0 = *(const v2f*)(a0p + kk);
    v2f a1 = *(const v2f*)(a1p + kk);
    v2f a2 = *(const v2f*)(a2p + kk);
    acc0 = __builtin_amdgcn_wmma_f32_16x16x4_f32(false, a0, false, b, (short)0,
                                                 acc0, false, false);
    acc1 = __builtin_amdgcn_wmma_f32_16x16x4_f32(false, a1, false, b, (short)0,
                                                 acc1, false, false);
    acc2 = __builtin_amdgcn_wmma_f32_16x16x4_f32(false, a2, false, b, (short)0,
                                                 acc2, false, false);
  }

  // C/D layout: VGPR i -> M = i (lanes 0-15) or i+8 (lanes 16-31), N = lane%16
  const int rb = m0 + ((lane < 16) ? 0 : 8);
#pragma unroll
  for (int i = 0; i < 8; ++i) {
    g.out[(size_t)(rb + i) * g.Nout + ncol] = acc0[i];
    g.out[(size_t)(rb + 16 + i) * g.Nout + ncol] = acc1[i];
    g.out[(size_t)(rb + 32 + i) * g.Nout + ncol] = acc2[i];
  }
}

// ---------------------------------------------------------------------------
// Launch
// ---------------------------------------------------------------------------
static inline int cdiv(size_t a, int b) { return (int)((a + b - 1) / b); }

extern "C" void kernel_launch(void* const* d_in, const int* in_sizes, int n_in,
                              void* d_out, int out_size, void* d_ws, size_t ws_size,
                              hipStream_t stream) {
  const float* edge_feats = (const float*)d_in[0];
  const float* c5_feats   = (const float*)d_in[1];
  const float* c6_feats   = (const float*)d_in[2];
  const int*   e_atoms    = (const int*)d_in[3];
  const int*   c5_atoms   = (const int*)d_in[4];
  const int*   c6_atoms   = (const int*)d_in[5];
  const float* e_W1 = (const float*)d_in[6];
  const float* e_g1 = (const float*)d_in[7];
  const float* e_b1 = (const float*)d_in[8];
  const float* e_W2 = (const float*)d_in[9];
  const float* e_g2 = (const float*)d_in[10];
  const float* e_b2 = (const float*)d_in[11];
  const float* c_W1 = (const float*)d_in[12];
  const float* c_g1 = (const float*)d_in[13];
  const float* c_b1 = (const float*)d_in[14];
  const float* c_W2 = (const float*)d_in[15];
  const float* c_g2 = (const float*)d_in[16];
  const float* c_b2 = (const float*)d_in[17];

  float* ws = (float*)d_ws;
  float* node64    = ws + OFF_NODE64;
  float* node128_5 = ws + OFF_NODE128_5;
  float* node128_6 = ws + OFF_NODE128_6;
  float* node320   = ws + OFF_NODE320;
  float* stats     = ws + OFF_STATS;
  int*   cnt5      = (int*)(ws + OFF_CNT5);
  int*   cnt6      = (int*)(ws + OFF_CNT6);
  float* bc64_5    = ws + OFF_BC64_5;
  float* bc64_6    = ws + OFF_BC64_6;
  float* bc128_5   = ws + OFF_BC128_5;
  float* bc128_6   = ws + OFF_BC128_6;
  float* h1        = ws + OFF_H1;
  float* sb        = ws + OFF_SB;
  float* wpk_e1    = ws + OFF_WPK_E1;
  float* wpk_e2    = ws + OFF_WPK_E2;
  float* wpk_c1    = ws + OFF_WPK_C1;
  float* wpk_c2    = ws + OFF_WPK_C2;

  float* out = (float*)d_out;
  float* out_e  = out;
  float* out_c5 = out + OUT_OFF_C5;
  float* out_c6 = out + OUT_OFF_C6;

  // ---- 0. zero accumulators / stats / counts; pack weights ----
  zero_f32x4<<<4096, 256, 0, stream>>>((float4*)ws, ZERO_COUNT / 4);
  pack_w<<<cdiv((size_t)704 * 128, 256), 256, 0, stream>>>(e_W1, wpk_e1, 704, 128);
  pack_w<<<cdiv((size_t)128 * 64, 256), 256, 0, stream>>>(e_W2, wpk_e2, 128, 64);
  pack_w<<<cdiv((size_t)320 * 128, 256), 256, 0, stream>>>(c_W1, wpk_c1, 320, 128);
  pack_w<<<cdiv((size_t)128 * 64, 256), 256, 0, stream>>>(c_W2, wpk_c2, 128, 64);

  // ---- 1. atom histograms + edge -> node64 ----
  count_atoms<<<cdiv(C5_ROWS, 256), 256, 0, stream>>>(c5_atoms, C5_ROWS, cnt5);
  count_atoms<<<cdiv(C6_ROWS, 256), 256, 0, stream>>>(c6_atoms, C6_ROWS, cnt6);
  scatter_node64<<<cdiv((size_t)E_ROWS * 64, 256), 256, 0, stream>>>(
      edge_feats, e_atoms, node64, E_ROWS);

  // ---- 2. first-gather broadcast tables ----
  seg_sum_gather4<<<cdiv((size_t)NC5 * 16, 256), 256, 0, stream>>>(
      (const float4*)node64, c5_atoms, (float4*)bc64_5, NC5, 16, 5);
  seg_sum_gather4<<<cdiv((size_t)NC6 * 16, 256), 256, 0, stream>>>(
      (const float4*)node64, c6_atoms, (float4*)bc64_6, NC6, 16, 6);

  // ---- 3. node128 = [cnt*node64 | scatter(bc64)] ----
  scatter_bc<<<cdiv((size_t)C5_ROWS * 64, 256), 256, 0, stream>>>(
      bc64_5, c5_atoms, node128_5, C5_ROWS, 5, 64, 128, 64);
  scatter_bc<<<cdiv((size_t)C6_ROWS * 64, 256), 256, 0, stream>>>(
      bc64_6, c6_atoms, node128_6, C6_ROWS, 6, 64, 128, 64);
  fill_node128_4<<<cdiv((size_t)NNODES * 16, 256), 256, 0, stream>>>(
      (const float4*)node64, cnt5, (float4*)node128_5);
  fill_node128_4<<<cdiv((size_t)NNODES * 16, 256), 256, 0, stream>>>(
      (const float4*)node64, cnt6, (float4*)node128_6);

  // ---- 4. second-gather broadcast tables ----
  seg_sum_gather4<<<cdiv((size_t)NC5 * 32, 256), 256, 0, stream>>>(
      (const float4*)node128_5, c5_atoms, (float4*)bc128_5, NC5, 32, 5);
  seg_sum_gather4<<<cdiv((size_t)NC6 * 32, 256), 256, 0, stream>>>(
      (const float4*)node128_6, c6_atoms, (float4*)bc128_6, NC6, 32, 6);

  // ---- 5. node320 ----
  scatter_bc<<<cdiv((size_t)C5_ROWS * 128, 256), 256, 0, stream>>>(
      bc128_5, c5_atoms, node320, C5_ROWS, 5, 128, 320, 128);
  scatter_bc<<<cdiv((size_t)C6_ROWS * 128, 256), 256, 0, stream>>>(
      bc128_6, c6_atoms, node320, C6_ROWS, 6, 128, 320, 128);
  scatter_feats<<<cdiv((size_t)C5_ROWS * 64, 256), 256, 0, stream>>>(
      c5_feats, c5_atoms, node320, C5_ROWS);
  scatter_feats<<<cdiv((size_t)C6_ROWS * 64, 256), 256, 0, stream>>>(
      c6_feats, c6_atoms, node320, C6_ROWS);
  fill_node320_4<<<cdiv((size_t)NNODES * 32, 256), 256, 0, stream>>>(
      (const float4*)node128_5, (const float4*)node128_6, cnt5, cnt6,
      (float4*)node320);

  const size_t smem_e1 = (size_t)16 * MT * (11 * H + 4) * 4;
  const size_t smem_c1 = (size_t)16 * MT * (5 * H + 4) * 4;
  const size_t smem_l2 = (size_t)16 * MT * (2 * H + 4) * 4;

  // =================== edge MLP ===================
  {
    GemmArgs g{};
    g.Wp = wpk_e1; g.out = h1; g.K = 11 * H; g.Nout = 2 * H;
    g.feats = edge_feats; g.node = node320; g.atoms = e_atoms; g.S = 2;
    gemm_wmma<GM_EDGE1><<<E_ROWS / (16 * MT), 256, smem_e1, stream>>>(g);
  }
  col_stats<<<512, 128, 0, stream>>>(h1, E_ROWS, 128, stats + SET_E1 * 256);
  bn_finalize<<<1, 128, 0, stream>>>(stats + SET_E1 * 256, e_g1, e_b1,
                                     sb + SET_E1 * 256, E_ROWS, 128);
  {
    GemmArgs g{};
    g.Wp = wpk_e2; g.out = out_e; g.K = 2 * H; g.Nout = H;
    g.feats = h1; g.sb = sb + SET_E1 * 256;
    gemm_wmma<GM_BNRELU><<<E_ROWS / (16 * MT), 128, smem_l2, stream>>>(g);
  }
  col_stats<<<512, 64, 0, stream>>>(out_e, E_ROWS, 64, stats + SET_E2 * 256);
  bn_finalize<<<1, 64, 0, stream>>>(stats + SET_E2 * 256, e_g2, e_b2,
                                    sb + SET_E2 * 256, E_ROWS, 64);
  bn_relu_apply4<<<cdiv((size_t)E_ROWS * 16, 256), 256, 0, stream>>>(
      (float4*)out_e, sb + SET_E2 * 256, (size_t)E_ROWS * 16, 64);

  // =================== cycle5 MLP ===================
  {
    GemmArgs g{};
    g.Wp = wpk_c1; g.out = h1; g.K = 5 * H; g.Nout = 2 * H;
    g.feats = c5_feats; g.node = node128_5; g.bcbuf = bc128_5;
    g.atoms = c5_atoms; g.S = 5;
    gemm_wmma<GM_CYC1><<<C5_ROWS / (16 * MT), 256, smem_c1, stream>>>(g);
  }
  col_stats<<<512, 128, 0, stream>>>(h1, C5_ROWS, 128, stats + SET_C51 * 256);
  bn_finalize<<<1, 128, 0, stream>>>(stats + SET_C51 * 256, c_g1, c_b1,
                                     sb + SET_C51 * 256, C5_ROWS, 128);
  {
    GemmArgs g{};
    g.Wp = wpk_c2; g.out = out_c5; g.K = 2 * H; g.Nout = H;
    g.feats = h1; g.sb = sb + SET_C51 * 256;
    gemm_wmma<GM_BNRELU><<<C5_ROWS / (16 * MT), 128, smem_l2, stream>>>(g);
  }
  col_stats<<<512, 64, 0, stream>>>(out_c5, C5_ROWS, 64, stats + SET_C52 * 256);
  bn_finalize<<<1, 64, 0, stream>>>(stats + SET_C52 * 256, c_g2, c_b2,
                                    sb + SET_C52 * 256, C5_ROWS, 64);
  bn_relu_apply4<<<cdiv((size_t)C5_ROWS * 16, 256), 256, 0, stream>>>(
      (float4*)out_c5, sb + SET_C52 * 256, (size_t)C5_ROWS * 16, 64);

  // =================== cycle6 MLP ===================
  {
    GemmArgs g{};
    g.Wp = wpk_c1; g.out = h1; g.K = 5 * H; g.Nout = 2 * H;
    g.feats = c6_feats; g.node = node128_6; g.bcbuf = bc128_6;
    g.atoms = c6_atoms; g.S = 6;
    gemm_wmma<GM_CYC1><<<C6_ROWS / (16 * MT), 256, smem_c1, stream>>>(g);
  }
  col_stats<<<512, 128, 0, stream>>>(h1, C6_ROWS, 128, stats + SET_C61 * 256);
  bn_finalize<<<1, 128, 0, stream>>>(stats + SET_C61 * 256, c_g1, c_b1,
                                     sb + SET_C61 * 256, C6_ROWS, 128);
  {
    GemmArgs g{};
    g.Wp = wpk_c2; g.out = out_c6; g.K = 2 * H; g.Nout = H;
    g.feats = h1; g.sb = sb + SET_C61 * 256;
    gemm_wmma<GM_BNRELU><<<C6_ROWS / (16 * MT), 128, smem_l2, stream>>>(g);
  }
  col_stats<<<512, 64, 0, stream>>>(out_c6, C6_ROWS, 64, stats + SET_C62 * 256);
  bn_finalize<<<1, 64, 0, stream>>>(stats + SET_C62 * 256, c_g2, c_b2,
                                    sb + SET_C62 * 256, C6_ROWS, 64);
  bn_relu_apply4<<<cdiv((size_t)C6_ROWS * 16, 256), 256, 0, stream>>>(
      (float4*)out_c6, sb + SET_C62 * 256, (size_t)C6_ROWS * 16, 64);
}